// SparseAttentionLayer_72378788873002
// MI455X (gfx1250) — compile-verified
//
#include <hip/hip_runtime.h>

typedef __attribute__((ext_vector_type(2))) float v2f;
typedef __attribute__((ext_vector_type(8))) float v8f;

#define Bb 256
#define Nn 64
#define Mm 64
#define Dd 128
#define Kk 10

__global__ __launch_bounds__(256) void sparse_attn_fused(
    const float* __restrict__ F,   // [B,N,D] featureVec
    const float* __restrict__ C,   // [B,M,D] contextVec
    const float* __restrict__ Wq,  // [D,D]
    const float* __restrict__ Wk,  // [D,D]
    float* __restrict__ out)       // [B, N*K, D]
{
    // 128 KB static LDS (CDNA5: up to 320 KB per workgroup -> 2 WGs/WGP resident)
    __shared__ float sF[Nn * Dd];   // featureVec tile   (kept for output phase)
    __shared__ float sC[Mm * Dd];   // contextVec tile   (kept for output phase)
    __shared__ float sQ[Nn * Dd];   // Q; reused for scores + topk tables
    __shared__ float sK[Mm * Dd];   // projected keys

    const int tid   = threadIdx.x;
    const int lane  = tid & 31;
    const int l15   = lane & 15;
    const int khalf = (lane >> 4) << 1;   // 0 or 2: which K-pair this half-wave holds
    const int mhalf = (lane >> 4) << 3;   // 0 or 8: C/D row offset for this half-wave
    const int wave  = tid >> 5;
    const int b     = blockIdx.x;

    // ---- Phase 1: stage F and C tiles into LDS (float4 vector loads) ----
    {
        const float4* gF = (const float4*)(F + (size_t)b * Nn * Dd);
        const float4* gC = (const float4*)(C + (size_t)b * Mm * Dd);
        float4* s4F = (float4*)sF;
        float4* s4C = (float4*)sC;
        for (int i = tid; i < Nn * Dd / 4; i += 256) {
            s4F[i] = gF[i];
            s4C[i] = gC[i];
        }
    }
    __syncthreads();

    // ---- Phase 2: projections  Q = F @ Wq^T,  Kp = C @ Wk^T  (WMMA f32 16x16x4) ----
    // wave & 1 selects which projection; wave >> 1 selects the 16-row tile.
    {
        const float* src = (wave & 1) ? sC : sF;
        const float* W   = (wave & 1) ? Wk : Wq;
        float*       dst = (wave & 1) ? sK : sQ;
        const int mbase  = (wave >> 1) * 16;

        v8f acc[8];
        #pragma unroll
        for (int nt = 0; nt < 8; ++nt) acc[nt] = {};

        for (int k0 = 0; k0 < Dd; k0 += 4) {
            // A fragment: rows mbase..mbase+15, K-slice k0..k0+3
            v2f a;
            {
                const float* ap = src + (mbase + l15) * Dd + k0 + khalf;
                a.x = ap[0]; a.y = ap[1];
            }
            #pragma unroll
            for (int nt = 0; nt < 8; ++nt) {
                // B fragment: B[k][n] = W[n][k]  (x @ W^T), W rows L2-resident
                v2f bf;
                const float* bp = W + (nt * 16 + l15) * Dd + k0 + khalf;
                bf.x = bp[0]; bf.y = bp[1];
                acc[nt] = __builtin_amdgcn_wmma_f32_16x16x4_f32(
                    false, a, false, bf, (short)0, acc[nt], false, false);
            }
        }
        // D tile layout: VGPR r -> row mbase + r + mhalf, col = nt*16 + l15
        #pragma unroll
        for (int nt = 0; nt < 8; ++nt)
            #pragma unroll
            for (int r = 0; r < 8; ++r)
                dst[(mbase + r + mhalf) * Dd + nt * 16 + l15] = acc[nt][r];
    }
    __syncthreads();

    // ---- Phase 3: scores = Q @ Kp^T / sqrt(D)  (64x64, 16 tiles, 2 per wave) ----
    v8f sacc0 = {}, sacc1 = {};
    const int smt  = wave >> 1;          // score M-tile 0..3
    const int snt0 = (wave & 1) * 2;     // score N-tile base 0 or 2
    for (int d0 = 0; d0 < Dd; d0 += 4) {
        v2f a;
        {
            const float* ap = sQ + (smt * 16 + l15) * Dd + d0 + khalf;
            a.x = ap[0]; a.y = ap[1];
        }
        v2f b0, b1;
        {
            const float* bp0 = sK + ((snt0 + 0) * 16 + l15) * Dd + d0 + khalf;
            const float* bp1 = sK + ((snt0 + 1) * 16 + l15) * Dd + d0 + khalf;
            b0.x = bp0[0]; b0.y = bp0[1];
            b1.x = bp1[0]; b1.y = bp1[1];
        }
        sacc0 = __builtin_amdgcn_wmma_f32_16x16x4_f32(false, a, false, b0, (short)0, sacc0, false, false);
        sacc1 = __builtin_amdgcn_wmma_f32_16x16x4_f32(false, a, false, b1, (short)0, sacc1, false, false);
    }
    __syncthreads();   // all WMMA reads of sQ complete -> safe to repurpose sQ

    float* sS  = sQ;                       // scores [64][64]   (16 KB)
    float* sTw = sQ + Nn * Mm;             // softmax weights [64][10]
    int*   sTi = (int*)(sQ + Nn * Mm + Nn * Kk);  // indices [64][10]
    const float scale = 0.08838834764831845f;     // 1/sqrt(128)

    #pragma unroll
    for (int r = 0; r < 8; ++r) {
        sS[(smt * 16 + r + mhalf) * Mm + (snt0 + 0) * 16 + l15] = sacc0[r] * scale;
        sS[(smt * 16 + r + mhalf) * Mm + (snt0 + 1) * 16 + l15] = sacc1[r] * scale;
    }
    __syncthreads();

    // ---- Phase 4: top-10 (lowest-index tie-break) + softmax, one thread per row ----
    if (tid < Nn) {
        float* row = sS + tid * Mm;
        float tv[Kk];
        int   ti[Kk];
        #pragma unroll
        for (int k = 0; k < Kk; ++k) {
            float best = -1e30f; int bi = 0;
            for (int m = 0; m < Mm; ++m) {
                float v = row[m];
                if (v > best) { best = v; bi = m; }   // strict '>' => lowest index on ties
            }
            tv[k] = best; ti[k] = bi;
            row[bi] = -1e30f;
        }
        float mx = tv[0], sum = 0.0f, e[Kk];
        #pragma unroll
        for (int k = 0; k < Kk; ++k) { e[k] = __expf(tv[k] - mx); sum += e[k]; }
        float inv = 1.0f / sum;
        #pragma unroll
        for (int k = 0; k < Kk; ++k) {
            sTw[tid * Kk + k] = e[k] * inv;
            sTi[tid * Kk + k] = ti[k];
        }
    }
    __syncthreads();

    // ---- Phase 5: out[b, n*K+k, d] = w[n,k] * F[n,d] * C[idx[n,k], d] ----
    // Vectorized float4: dominant traffic phase (84 MB of ~100 MB total).
    // ds_load_b128 x2 + global_store_b128 per 16 output bytes, fully coalesced.
    {
        float4* ob4 = (float4*)(out + (size_t)b * (Nn * Kk * Dd));
        const float4* s4F = (const float4*)sF;
        const float4* s4C = (const float4*)sC;
        const int D4 = Dd / 4;                       // 32 float4 per row
        for (int e4 = tid; e4 < Nn * Kk * D4; e4 += 256) {
            int n  = e4 / (Kk * D4);
            int r  = e4 - n * (Kk * D4);
            int k  = r / D4;
            int d4 = r - k * D4;
            int idx = sTi[n * Kk + k];
            float w = sTw[n * Kk + k];
            float4 f = s4F[n * D4 + d4];
            float4 c = s4C[idx * D4 + d4];
            float4 o;
            o.x = w * f.x * c.x;
            o.y = w * f.y * c.y;
            o.z = w * f.z * c.z;
            o.w = w * f.w * c.w;
            ob4[e4] = o;
        }
    }
}

extern "C" void kernel_launch(void* const* d_in, const int* in_sizes, int n_in,
                              void* d_out, int out_size, void* d_ws, size_t ws_size,
                              hipStream_t stream) {
    (void)in_sizes; (void)n_in; (void)out_size; (void)d_ws; (void)ws_size;
    const float* F  = (const float*)d_in[0];
    const float* C  = (const float*)d_in[1];
    const float* Wq = (const float*)d_in[2];
    const float* Wk = (const float*)d_in[3];
    float* out = (float*)d_out;
    sparse_attn_fused<<<Bb, 256, 0, stream>>>(F, C, Wq, Wk, out);
}